// Quadratics_56994216017961
// MI455X (gfx1250) — compile-verified
//
#include <hip/hip_runtime.h>
#include <hip/hip_bf16.h>
#include <math.h>

typedef __attribute__((ext_vector_type(2))) float v2f;
typedef __attribute__((ext_vector_type(8))) float v8f;

#define RED_BLOCKS 512
#define RED_THREADS 256
#define EPS_F 1e-24f

// ---------------------------------------------------------------------------
// Kernel 1: per-block partial sums of |g|^2 and |x_curr - x_prev|^2
// ---------------------------------------------------------------------------
__global__ void lopt_reduce_partial(const float* __restrict__ grad,
                                    const float* __restrict__ xp,
                                    const float* __restrict__ xc,
                                    float* __restrict__ partials, int n) {
  __shared__ float sh[2 * RED_THREADS];
  float sg = 0.f, sm = 0.f;
  int tid = threadIdx.x;
  for (int i = blockIdx.x * blockDim.x + tid; i < n; i += gridDim.x * blockDim.x) {
    float gv = grad[i];
    float mv = xc[i] - xp[i];
    sg = fmaf(gv, gv, sg);
    sm = fmaf(mv, mv, sm);
  }
  sh[tid] = sg;
  sh[RED_THREADS + tid] = sm;
  __syncthreads();
  for (int s = RED_THREADS / 2; s > 0; s >>= 1) {
    if (tid < s) {
      sh[tid] += sh[tid + s];
      sh[RED_THREADS + tid] += sh[RED_THREADS + tid + s];
    }
    __syncthreads();
  }
  if (tid == 0) {
    partials[2 * blockIdx.x] = sh[0];
    partials[2 * blockIdx.x + 1] = sh[RED_THREADS];
  }
}

// ---------------------------------------------------------------------------
// Kernel 2: finalize norms + scalar coefficient MLP -> {inv_gn, inv_mn, step}
// ---------------------------------------------------------------------------
__global__ void lopt_finalize(const float* __restrict__ partials, int nb,
                              const float* __restrict__ loss_curr,
                              const float* __restrict__ loss_prev,
                              const float* __restrict__ Wl1,
                              const float* __restrict__ Wl2,
                              const float* __restrict__ Wl3,
                              const float* __restrict__ Wl4,
                              const float* __restrict__ Wl5,
                              float* __restrict__ scal) {
  __shared__ float sh[2 * RED_THREADS];
  int tid = threadIdx.x;
  float sg = 0.f, sm = 0.f;
  for (int i = tid; i < nb; i += RED_THREADS) {
    sg += partials[2 * i];
    sm += partials[2 * i + 1];
  }
  sh[tid] = sg;
  sh[RED_THREADS + tid] = sm;
  __syncthreads();
  for (int s = RED_THREADS / 2; s > 0; s >>= 1) {
    if (tid < s) {
      sh[tid] += sh[tid + s];
      sh[RED_THREADS + tid] += sh[RED_THREADS + tid + s];
    }
    __syncthreads();
  }
  if (tid == 0) {
    float gn = sqrtf(sh[0]);
    float mn = sqrtf(sh[RED_THREADS]);
    float f[4] = { log1pf(gn), log1pf(mn), log1pf(loss_curr[0]), log1pf(loss_prev[0]) };
    float h1[8], h2[8], h3[8], h4[8];
    for (int o = 0; o < 8; ++o) {
      float a = 0.f;
      for (int k = 0; k < 4; ++k) a = fmaf(Wl1[o * 4 + k], f[k], a);
      h1[o] = fmaxf(a, 0.f);
    }
    for (int o = 0; o < 8; ++o) {
      float a = 0.f;
      for (int k = 0; k < 8; ++k) a = fmaf(Wl2[o * 8 + k], h1[k], a);
      h2[o] = fmaxf(a, 0.f);
    }
    for (int o = 0; o < 8; ++o) {
      float a = 0.f;
      for (int k = 0; k < 8; ++k) a = fmaf(Wl3[o * 8 + k], h2[k], a);
      h3[o] = fmaxf(a, 0.f);
    }
    for (int o = 0; o < 8; ++o) {
      float a = 0.f;
      for (int k = 0; k < 8; ++k) a = fmaf(Wl4[o * 8 + k], h3[k], a);
      h4[o] = a;  // no relu on 4th layer
    }
    float step = 0.f;
    for (int k = 0; k < 8; ++k) step = fmaf(Wl5[k], h4[k], step);
    scal[0] = (gn > EPS_F) ? (1.0f / gn) : 1.0f;
    scal[1] = (mn > EPS_F) ? (1.0f / mn) : 1.0f;
    scal[2] = step;
  }
}

// ---------------------------------------------------------------------------
// Kernel 3: direction network via V_WMMA_F32_16X16X4_F32
// ---------------------------------------------------------------------------
__device__ __forceinline__ float lopt_bpermf(int byte_idx, float v) {
  return __int_as_float(__builtin_amdgcn_ds_bpermute(byte_idx, __float_as_int(v)));
}

// single-instruction ReLU (avoid fmaxf's extra canonicalize v_max)
__device__ __forceinline__ float lopt_relu(float x) {
  float y;
  asm("v_max_num_f32 %0, %1, 0" : "=v"(y) : "v"(x));
  return y;
}

// one 10x10 layer: D-layout activations -> B fragments -> 3 chained WMMAs
__device__ __forceinline__ v8f lopt_layer10(const v2f A0, const v2f A1, const v2f A2,
                                            v8f din, int idx_lo, int idx_hi, int hiq) {
  v2f b0, b1, b2;
  // chunk 0: K = 2*hi + v  (channels 0..3), source lanes 0..15, reg = K
  b0.x = lopt_bpermf(idx_lo, hiq ? din[2] : din[0]);
  b0.y = lopt_bpermf(idx_lo, hiq ? din[3] : din[1]);
  // chunk 1: K = 4 + 2*hi + v (channels 4..7), source lanes 0..15, reg = K
  b1.x = lopt_bpermf(idx_lo, hiq ? din[6] : din[4]);
  b1.y = lopt_bpermf(idx_lo, hiq ? din[7] : din[5]);
  // chunk 2: K = 8 + 2*hi + v (channels 8..9 valid only in lower half; 10..11 = 0)
  float t0 = lopt_bpermf(idx_hi, din[0]);
  float t1 = lopt_bpermf(idx_hi, din[1]);
  b2.x = hiq ? 0.0f : t0;
  b2.y = hiq ? 0.0f : t1;
  v8f d = {};
  d = __builtin_amdgcn_wmma_f32_16x16x4_f32(false, A0, false, b0, (short)0, d, false, false);
  d = __builtin_amdgcn_wmma_f32_16x16x4_f32(false, A1, false, b1, (short)0, d, false, false);
  d = __builtin_amdgcn_wmma_f32_16x16x4_f32(false, A2, false, b2, (short)0, d, false, false);
  return d;
}

__device__ __forceinline__ float lopt_wpad(const float* W, int rows, int cols, int m, int k) {
  return (m < rows && k < cols) ? W[m * cols + k] : 0.0f;
}

__global__ void lopt_direction(const float* __restrict__ grad,
                               const float* __restrict__ xp,
                               const float* __restrict__ xc,
                               const float* __restrict__ Wc1,
                               const float* __restrict__ Wc2,
                               const float* __restrict__ Wc3,
                               const float* __restrict__ Wc4,
                               const float* __restrict__ Wc5,
                               const float* __restrict__ Wc6,
                               const float* __restrict__ Wc7,
                               const float* __restrict__ scal,
                               float* __restrict__ out, int dim) {
  const int lane = threadIdx.x & 31;
  const int hiq  = lane >> 4;   // 0 = lanes 0-15, 1 = lanes 16-31
  const int col  = lane & 15;   // N index within tile / M row for A fragments

  const float inv_gn = scal[0];
  const float inv_mn = scal[1];
  const float step   = scal[2];

  // ---- weight fragments in A layout (lane: M=col, reg v: K = 4t+2*hi+v) ----
  v2f a1;  // Wc1: 10x3, single K-chunk (K=3 padded 0)
  a1.x = lopt_wpad(Wc1, 10, 3, col, 2 * hiq + 0);
  a1.y = lopt_wpad(Wc1, 10, 3, col, 2 * hiq + 1);

  const float* Ws[5] = { Wc2, Wc3, Wc4, Wc5, Wc6 };
  v2f A[5][3];
#pragma unroll
  for (int l = 0; l < 5; ++l) {
#pragma unroll
    for (int t = 0; t < 3; ++t) {
      A[l][t].x = lopt_wpad(Ws[l], 10, 10, col, 4 * t + 2 * hiq + 0);
      A[l][t].y = lopt_wpad(Ws[l], 10, 10, col, 4 * t + 2 * hiq + 1);
    }
  }
  // Wc7 (1x10) as VALU dot over D-layout channels r + 8*hi
  float w7[8];
#pragma unroll
  for (int r = 0; r < 8; ++r) {
    int c = r + 8 * hiq;
    w7[r] = (c < 10) ? Wc7[c] : 0.0f;
  }

  const int idx_lo  = col << 2;          // gather from lane (col)
  const int idx_hi2 = (col + 16) << 2;   // gather from lane (col + 16)
  const int idx_x16 = (lane ^ 16) << 2;  // cross-half partner

  const int ntiles = dim >> 4;
  const int wave   = (blockIdx.x * blockDim.x + threadIdx.x) >> 5;
  const int nwaves = (gridDim.x * blockDim.x) >> 5;

  for (int tile = wave; tile < ntiles; tile += nwaves) {
    const int n = (tile << 4) + col;   // both lane-halves mirror the 16 columns

    // prefetch next grid-stride tile (wave-uniform branch; EXEC stays full)
    const int tnext = tile + nwaves;
    if (tnext < ntiles) {
      const int nn = (tnext << 4) + col;
      __builtin_prefetch(&grad[nn], 0, 0);
      __builtin_prefetch(&xp[nn], 0, 0);
      __builtin_prefetch(&xc[nn], 0, 0);
    }

    const float xcn = xc[n];
    const float gv  = grad[n] * inv_gn;
    const float mv  = (xcn - xp[n]) * inv_mn;

    // input B fragment for Wc1: K=0:g, K=1:m, K=2:g*m, K=3:0
    v2f b;
    b.x = hiq ? gv * mv : gv;
    b.y = hiq ? 0.0f    : mv;
    v8f d = {};
    d = __builtin_amdgcn_wmma_f32_16x16x4_f32(false, a1, false, b, (short)0, d, false, false);
    // d = Wc1 @ x   (no activation)

#pragma unroll
    for (int l = 0; l < 5; ++l) {
      d = lopt_layer10(A[l][0], A[l][1], A[l][2], d, idx_lo, idx_hi2, hiq);
      if (l < 4) {  // relu after Wc2..Wc5, not after Wc6
#pragma unroll
        for (int r = 0; r < 8; ++r) d[r] = lopt_relu(d[r]);
      }
    }

    // Wc7: dir[n] = sum_c Wc7[c] * act[c][n]; each half holds 8 channels
    float p = 0.f;
#pragma unroll
    for (int r = 0; r < 8; ++r) p = fmaf(w7[r], d[r], p);
    float dir = p + lopt_bpermf(idx_x16, p);

    if (!hiq) out[n] = fmaf(-step, dir, xcn);
  }
}

// ---------------------------------------------------------------------------
extern "C" void kernel_launch(void* const* d_in, const int* in_sizes, int n_in,
                              void* d_out, int out_size, void* d_ws, size_t ws_size,
                              hipStream_t stream) {
  const float* grad = (const float*)d_in[0];
  const float* xp   = (const float*)d_in[1];
  const float* xc   = (const float*)d_in[2];
  const float* lc   = (const float*)d_in[3];
  const float* lp   = (const float*)d_in[4];
  const float* Wc1  = (const float*)d_in[5];
  const float* Wc2  = (const float*)d_in[6];
  const float* Wc3  = (const float*)d_in[7];
  const float* Wc4  = (const float*)d_in[8];
  const float* Wc5  = (const float*)d_in[9];
  const float* Wc6  = (const float*)d_in[10];
  const float* Wc7  = (const float*)d_in[11];
  const float* Wl1  = (const float*)d_in[12];
  const float* Wl2  = (const float*)d_in[13];
  const float* Wl3  = (const float*)d_in[14];
  const float* Wl4  = (const float*)d_in[15];
  const float* Wl5  = (const float*)d_in[16];

  const int dim = in_sizes[0];
  float* ws       = (float*)d_ws;
  float* partials = ws;                      // 2 * RED_BLOCKS floats
  float* scal     = ws + 2 * RED_BLOCKS;     // {inv_gn, inv_mn, step}

  lopt_reduce_partial<<<RED_BLOCKS, RED_THREADS, 0, stream>>>(grad, xp, xc, partials, dim);
  lopt_finalize<<<1, RED_THREADS, 0, stream>>>(partials, RED_BLOCKS, lc, lp,
                                               Wl1, Wl2, Wl3, Wl4, Wl5, scal);
  lopt_direction<<<2048, 256, 0, stream>>>(grad, xp, xc,
                                           Wc1, Wc2, Wc3, Wc4, Wc5, Wc6, Wc7,
                                           scal, (float*)d_out, dim);
}